// Seq2matCMOW_21260088115487
// MI455X (gfx1250) — compile-verified
//
#include <hip/hip_runtime.h>

// CDNA5 / gfx1250: fp32 matrix prefix scan via V_WMMA_F32_16X16X4_F32.
// Reference: all_outputs[b, i] = X[b,0] @ X[b,1] @ ... @ X[b,i]  (fp32)
//            x[b] = all_outputs[b, SEQ-1]
// d_out layout: [x : 32*128*128 floats][all_outputs : 32*256*128*128 floats]
//
// Strategy (when ws_size >= 32 MB): work-efficient 3-phase scan
//   1) local_scan   : 512 blocks, chunk-local prefix products (serial depth 15)
//   2) offset_scan  : 32 blocks, exclusive scan of chunk totals -> d_ws offsets
//   3) recombine    : 480 independent blocks, out = off[c] @ local (in place)
// Fallback: single-kernel sequential scan (32 blocks, serial depth 255).

typedef __attribute__((ext_vector_type(2))) float v2f;
typedef __attribute__((ext_vector_type(8))) float v8f;

constexpr int kD       = 128;
constexpr int kSeq     = 256;
constexpr int kBsz     = 32;
constexpr int kChunk   = 16;                 // matrices per chunk
constexpr int kNChunks = kSeq / kChunk;      // 16

// ---- one 128x128x128 matmul "step": acc(4 tiles) += A(LDS) @ B(global) ----
// A-fragment (32-bit A layout): lane<16 holds K=k0,k0+1; lane>=16 K=k0+2,k0+3.
// B-fragment: VGPR v holds K=k0+v+2*lh, N=l16 within tile.
__device__ __forceinline__ void mm_step(const float* __restrict__ Alds,
                                        const float* __restrict__ Bglb,
                                        int arow, int lh, int l16, int tileN0,
                                        v8f acc[4])
{
    #pragma unroll 4
    for (int kk = 0; kk < kD / 4; ++kk) {
        const int k0 = kk * 4;
        v2f a = *(const v2f*)&Alds[arow * kD + k0 + 2 * lh];
        const float* bbase = Bglb + (size_t)(k0 + 2 * lh) * kD + l16;
        #pragma unroll
        for (int t = 0; t < 4; ++t) {
            const float* p = bbase + (tileN0 + t) * 16;
            v2f bf;
            bf.x = p[0];
            bf.y = p[kD];
            acc[t] = __builtin_amdgcn_wmma_f32_16x16x4_f32(
                false, a, false, bf, (short)0, acc[t], false, false);
        }
    }
}

// D layout: VGPR j -> row tileM*16 + j + 8*lh, col tileN*16 + l16.
// WITH_LDS is a compile-time flag so no per-store branches are emitted.
template <bool WITH_LDS>
__device__ __forceinline__ void store_tiles(float* __restrict__ gdst,
                                            float* __restrict__ lds_dst,
                                            const v8f acc[4],
                                            int tileM, int lh, int l16, int tileN0)
{
    #pragma unroll
    for (int t = 0; t < 4; ++t) {
        const int col = (tileN0 + t) * 16 + l16;
        #pragma unroll
        for (int j = 0; j < 8; ++j) {
            const int row = tileM * 16 + 8 * lh + j;
            float v = acc[t][j];
            if constexpr (WITH_LDS) lds_dst[row * kD + col] = v;
            gdst[row * kD + col] = v;
        }
    }
}

// ---------------- Phase 1: chunk-local prefix scans ----------------
__global__ __launch_bounds__(512, 1)
void local_scan(const float* __restrict__ x, float* __restrict__ out_all)
{
    __shared__ float P[kD * kD];   // running chunk-local prefix, 64 KB
    const int tid = threadIdx.x, wave = tid >> 5, lane = tid & 31;
    const int l16 = lane & 15, lh = lane >> 4;
    const int c = blockIdx.x, b = blockIdx.y;

    const float* xc = x       + ((size_t)b * kSeq + c * kChunk) * kD * kD;
    float*       oc = out_all + ((size_t)b * kSeq + c * kChunk) * kD * kD;

    for (int i = tid; i < kD * kD / 4; i += 512) {
        float4 v = ((const float4*)xc)[i];
        ((float4*)P)[i]  = v;
        ((float4*)oc)[i] = v;
    }
    __syncthreads();

    const int tileM = wave & 7, tileN0 = (wave >> 3) << 2, arow = tileM * 16 + l16;

    for (int s = 1; s < kChunk; ++s) {
        const float* xs = xc + (size_t)s * kD * kD;
        float*       os = oc + (size_t)s * kD * kD;
        if (s + 1 < kChunk) __builtin_prefetch(xs + kD * kD + tid * 32, 0, 3);

        v8f acc[4] = {};
        mm_step(P, xs, arow, lh, l16, tileN0, acc);
        __syncthreads();                       // old P fully read
        store_tiles<true>(os, P, acc, tileM, lh, l16, tileN0);
        __syncthreads();                       // new P visible
    }
}

// ---------------- Phase 2: exclusive scan of chunk totals -> offsets in ws ----
// off[1] = total[0];  off[c] = off[c-1] @ total[c-1]   (totals = local prefix @ pos 15)
__global__ __launch_bounds__(512, 1)
void offset_scan(const float* __restrict__ out_all, float* __restrict__ ws)
{
    __shared__ float O[kD * kD];
    const int tid = threadIdx.x, wave = tid >> 5, lane = tid & 31;
    const int l16 = lane & 15, lh = lane >> 4;
    const int b = blockIdx.x;

    const float* base = out_all + (size_t)b * kSeq * kD * kD;
    float*       wsb  = ws      + (size_t)b * kNChunks * kD * kD;

    // O = total[0]; ws[b][1] = O
    const float* t0 = base + (size_t)(kChunk - 1) * kD * kD;
    float*       w1 = wsb + (size_t)1 * kD * kD;
    for (int i = tid; i < kD * kD / 4; i += 512) {
        float4 v = ((const float4*)t0)[i];
        ((float4*)O)[i]  = v;
        ((float4*)w1)[i] = v;
    }
    __syncthreads();

    const int tileM = wave & 7, tileN0 = (wave >> 3) << 2, arow = tileM * 16 + l16;

    for (int c = 2; c < kNChunks; ++c) {
        const float* tc = base + (size_t)((c - 1) * kChunk + (kChunk - 1)) * kD * kD;
        float*       wc = wsb + (size_t)c * kD * kD;
        v8f acc[4] = {};
        mm_step(O, tc, arow, lh, l16, tileN0, acc);
        __syncthreads();
        store_tiles<true>(wc, O, acc, tileM, lh, l16, tileN0);
        __syncthreads();
    }
}

// ---------------- Phase 3: recombine (in place), fully parallel over chunks ----
__global__ __launch_bounds__(512, 1)
void recombine(float* __restrict__ out_all, const float* __restrict__ ws,
               float* __restrict__ out_last)
{
    __shared__ float O[kD * kD];   // offset matrix for this chunk (constant)
    const int tid = threadIdx.x, wave = tid >> 5, lane = tid & 31;
    const int l16 = lane & 15, lh = lane >> 4;
    const int c = blockIdx.x + 1;          // chunks 1..15 (chunk 0 already final)
    const int b = blockIdx.y;

    const float* off = ws + ((size_t)b * kNChunks + c) * kD * kD;
    float*       oc  = out_all + ((size_t)b * kSeq + c * kChunk) * kD * kD;

    for (int i = tid; i < kD * kD / 4; i += 512)
        ((float4*)O)[i] = ((const float4*)off)[i];
    __syncthreads();

    const int tileM = wave & 7, tileN0 = (wave >> 3) << 2, arow = tileM * 16 + l16;
    const bool last_chunk = (c == kNChunks - 1);

    for (int j = 0; j < kChunk; ++j) {
        float* mj = oc + (size_t)j * kD * kD;
        if (j + 1 < kChunk) __builtin_prefetch(mj + kD * kD + tid * 32, 0, 3);

        v8f acc[4] = {};
        mm_step(O, mj, arow, lh, l16, tileN0, acc);   // reads local[j]
        __syncthreads();                              // all waves done reading mj
        store_tiles<false>(mj, nullptr, acc, tileM, lh, l16, tileN0);  // in place
        if (last_chunk && j == kChunk - 1)
            store_tiles<false>(out_last + (size_t)b * kD * kD, nullptr, acc,
                               tileM, lh, l16, tileN0);
    }
}

// ---------------- Fallback: single-kernel sequential scan ----------------
__global__ __launch_bounds__(512, 1)
void prefix_matmul_scan(const float* __restrict__ x,
                        float* __restrict__ out_last,
                        float* __restrict__ out_all)
{
    __shared__ float P[kD * kD];
    const int tid = threadIdx.x, wave = tid >> 5, lane = tid & 31;
    const int l16 = lane & 15, lh = lane >> 4;
    const int b = blockIdx.x;
    const float* xb = x       + (size_t)b * kSeq * kD * kD;
    float*       ob = out_all + (size_t)b * kSeq * kD * kD;

    for (int i = tid; i < kD * kD / 4; i += 512) {
        float4 v = ((const float4*)xb)[i];
        ((float4*)P)[i]  = v;
        ((float4*)ob)[i] = v;
    }
    __syncthreads();

    const int tileM = wave & 7, tileN0 = (wave >> 3) << 2, arow = tileM * 16 + l16;

    for (int s = 1; s < kSeq; ++s) {
        const float* xs = xb + (size_t)s * kD * kD;
        float*       os = ob + (size_t)s * kD * kD;
        if (s + 1 < kSeq) __builtin_prefetch(xs + kD * kD + tid * 32, 0, 3);

        v8f acc[4] = {};
        mm_step(P, xs, arow, lh, l16, tileN0, acc);
        __syncthreads();
        store_tiles<true>(os, P, acc, tileM, lh, l16, tileN0);
        __syncthreads();
    }

    float* xl = out_last + (size_t)b * kD * kD;
    for (int i = tid; i < kD * kD / 4; i += 512)
        ((float4*)xl)[i] = ((const float4*)P)[i];
}

extern "C" void kernel_launch(void* const* d_in, const int* in_sizes, int n_in,
                              void* d_out, int out_size, void* d_ws, size_t ws_size,
                              hipStream_t stream) {
    (void)in_sizes; (void)n_in; (void)out_size;
    const float* x = (const float*)d_in[0];
    float* out      = (float*)d_out;
    float* out_last = out;                                   // 32*128*128 floats
    float* out_all  = out + (size_t)kBsz * kD * kD;          // 32*256*128*128 floats

    constexpr size_t kWsNeed =
        (size_t)kBsz * kNChunks * kD * kD * sizeof(float);   // 32 MB of offsets

    if (ws_size >= kWsNeed) {
        // Work-efficient 3-phase scan: fills the machine, serial depth 15+14+16.
        local_scan <<<dim3(kNChunks, kBsz),     dim3(512), 0, stream>>>(x, out_all);
        offset_scan<<<dim3(kBsz),               dim3(512), 0, stream>>>(out_all, (float*)d_ws);
        recombine  <<<dim3(kNChunks - 1, kBsz), dim3(512), 0, stream>>>(out_all, (const float*)d_ws, out_last);
    } else {
        // Fallback: sequential scan, no workspace required.
        prefix_matmul_scan<<<dim3(kBsz), dim3(512), 0, stream>>>(x, out_last, out_all);
    }
}